// UniformMatcher_28140625723714
// MI455X (gfx1250) — compile-verified
//
#include <hip/hip_runtime.h>
#include <stdint.h>

// ---------------------------------------------------------------------------
// UniformMatcher cost kernel for MI455X (gfx1250).
// out = [C (131072x512) | C1 (131072x512)] fp32, store-bandwidth-bound.
// CDNA5 paths: TDM tensor_load_to_lds staging of tgt boxes, s_wait_tensorcnt,
// LDS b128 reads, non-temporal global_store_b128 output stream.
// ---------------------------------------------------------------------------

typedef float         v4f  __attribute__((ext_vector_type(4)));
typedef unsigned int  u32x4 __attribute__((ext_vector_type(4)));
typedef int           i32x4 __attribute__((ext_vector_type(4)));
typedef int           i32x8 __attribute__((ext_vector_type(8)));

#define NQ_TOTAL (8 * 16384)   // 131072 rows total
#define T_BOXES  512           // targets
#define ROWS_PER_BLOCK 64
#define THREADS 256            // 8 wave32 = 2 stripes of 128 lanes over t

__global__ __launch_bounds__(THREADS)
void uniform_matcher_kernel(const float* __restrict__ pre_boxes,
                            const float* __restrict__ anchors,
                            const float* __restrict__ tgt,
                            float* __restrict__ out)
{
    __shared__ __align__(16) float s_tgt[T_BOXES * 4];  // raw xyxy, 8 KB

    const int tid = threadIdx.x;

    // ---- Stage tgt_bbox (8 KB) into LDS via the Tensor Data Mover ----
    // Only wave 0 reaches the TDM op (scalar branch; EXEC all-1s inside).
    if (tid < 32) {
        uint64_t ga  = (uint64_t)(uintptr_t)tgt;
        uint32_t lds = (uint32_t)(uintptr_t)(&s_tgt[0]); // LDS offset = flat[31:0]

        // D# group 0: count=1 | lds_addr | global_addr[56:0] | type=2
        u32x4 g0;
        g0[0] = 1u;                                   // count=1, is_restore=0
        g0[1] = lds;                                  // lds_addr (bytes)
        g0[2] = (uint32_t)ga;                         // global_addr[31:0]
        g0[3] = (uint32_t)(ga >> 32) | (2u << 30);    // addr[56:32] | type=2

        // D# group 1: 1-D tensor of 2048 fp32 (data_size=4B), tile = whole row
        i32x8 g1;
        g1[0] = (int)(2u << 16);      // wg_mask=0, data_size=2 (4 bytes)
        g1[1] = (int)(2048u << 16);   // tensor_dim0[15:0]=2048 at bits[63:48->31:16]
        g1[2] = (int)(1u << 16);      // tensor_dim0[31:16]=0, tensor_dim1=1
        g1[3] = (int)(2048u << 16);   // tile_dim0 = 2048
        g1[4] = 1;                    // tile_dim1 = 1, tile_dim2 = 0
        g1[5] = 2048;                 // tensor_dim0_stride
        g1[6] = 0;
        g1[7] = 0;

        i32x4 gz = {0, 0, 0, 0};
#if defined(__clang_major__) && (__clang_major__ >= 23)
        i32x8 gz8 = {0, 0, 0, 0, 0, 0, 0, 0};
        __builtin_amdgcn_tensor_load_to_lds(g0, g1, gz, gz, gz8, 0);
#else
        __builtin_amdgcn_tensor_load_to_lds(g0, g1, gz, gz, 0);
#endif
        __builtin_amdgcn_s_wait_tensorcnt(0);         // TENSORcnt == 0
    }
    __syncthreads();                                  // publish LDS to all waves

    // ---- Each lane owns 4 consecutive targets; cxcywh kept in 16 VGPRs ----
    const int stripe = tid >> 7;        // 0 or 1 (row interleave)
    const int tbase  = (tid & 127) * 4; // t column base

    float cx[4], cy[4], cw[4], ch[4];
#pragma unroll
    for (int j = 0; j < 4; ++j) {
        v4f tb = *(const v4f*)(&s_tgt[(tbase + j) * 4]);   // ds_load_b128
        cx[j] = (tb.x + tb.z) * 0.5f;
        cy[j] = (tb.y + tb.w) * 0.5f;
        cw[j] =  tb.z - tb.x;
        ch[j] =  tb.w - tb.y;
    }

    const long base_row = (long)blockIdx.x * ROWS_PER_BLOCK;
    float* __restrict__ outC  = out;
    float* __restrict__ outC1 = out + (long)NQ_TOTAL * T_BOXES;

    for (int r = stripe; r < ROWS_PER_BLOCK; r += 2) {
        const long row = base_row + r;
        // One cacheline per stripe; hardware broadcasts to all 128 lanes.
        v4f p = *(const v4f*)(pre_boxes + row * 4);        // global_load_b128
        v4f a = *(const v4f*)(anchors  + row * 4);

        const float pcx = (p.x + p.z) * 0.5f, pcy = (p.y + p.w) * 0.5f;
        const float pw  =  p.z - p.x,         ph  =  p.w - p.y;
        const float acx = (a.x + a.z) * 0.5f, acy = (a.y + a.w) * 0.5f;
        const float aw  =  a.z - a.x,         ah  =  a.w - a.y;

        v4f c, c1;
#pragma unroll
        for (int j = 0; j < 4; ++j) {
            // 4 subs + 3 adds with |abs| input modifiers per output element
            c[j]  = (fabsf(pcx - cx[j]) + fabsf(pcy - cy[j]))
                  + (fabsf(pw  - cw[j]) + fabsf(ph  - ch[j]));
            c1[j] = (fabsf(acx - cx[j]) + fabsf(acy - cy[j]))
                  + (fabsf(aw  - cw[j]) + fabsf(ah  - ch[j]));
        }

        // Streaming output (512 MiB >> L2): non-temporal b128 stores,
        // fully coalesced (128 lanes x 16 B = 2 KB contiguous per stripe).
        __builtin_nontemporal_store(c,  (v4f*)(outC  + row * T_BOXES + tbase));
        __builtin_nontemporal_store(c1, (v4f*)(outC1 + row * T_BOXES + tbase));
    }
}

extern "C" void kernel_launch(void* const* d_in, const int* in_sizes, int n_in,
                              void* d_out, int out_size, void* d_ws, size_t ws_size,
                              hipStream_t stream) {
    const float* pre = (const float*)d_in[0];   // [8,16384,4] fp32
    const float* anc = (const float*)d_in[1];   // [8,16384,4] fp32
    const float* tgt = (const float*)d_in[2];   // [512,4]     fp32
    float* out = (float*)d_out;                 // [2, 131072, 512] fp32

    dim3 grid(NQ_TOTAL / ROWS_PER_BLOCK);       // 2048 blocks
    uniform_matcher_kernel<<<grid, THREADS, 0, stream>>>(pre, anc, tgt, out);
}